// LAPSolverModule_63307817943835
// MI455X (gfx1250) — compile-verified
//
#include <hip/hip_runtime.h>
#include <hip/hip_bf16.h>

#define LAP_N    256
#define LAP_N1   257
#define LAP_K    9            // ceil(257/32) columns owned per lane
#define LAP_BIG  1e9f

// ---- LDS layout (bytes) ----
#define OFF_C    0                               // 256*256 f32 cost matrix
#define OFF_U    (LAP_N * LAP_N * 4)             // 262144 : u[257] row duals
#define OFF_P    (OFF_U + LAP_N1 * 4)            // 263172 : p[257] col->row match
#define OFF_WAY  (OFF_P + LAP_N1 * 4)            // 264200 : way[288]
#define SMEM_BYTES (OFF_WAY + 32 * LAP_K * 4)    // 265352 B  ( < 320KB WGP LDS )

typedef __attribute__((ext_vector_type(4))) unsigned int u32x4;
typedef __attribute__((ext_vector_type(8))) unsigned int u32x8;

// ---------------------------------------------------------------------------
// Exact Jonker-Volgenant LAP, one wave32, everything in LDS/registers.
// Cost matrix staged by ONE Tensor-Data-Mover op (tensor_load_to_lds).
// (Defined first so the disasm snippet shows the TDM op + hot loop.)
// ---------------------------------------------------------------------------
__global__ __launch_bounds__(32) void lap_solve(const float* __restrict__ C,
                                                float* __restrict__ out) {
  extern __shared__ unsigned char smem[];
  float* Cl  = reinterpret_cast<float*>(smem + OFF_C);
  float* u   = reinterpret_cast<float*>(smem + OFF_U);
  int*   p   = reinterpret_cast<int*>  (smem + OFF_P);
  int*   way = reinterpret_cast<int*>  (smem + OFF_WAY);

  const int lane = threadIdx.x;

  // ---- TDM: one descriptor moves the whole 256x256 f32 tile Global -> LDS.
  // D# group0: count=1 | lds_addr | global_addr[56:0] | type=2
  // D# group1: data_size=4B, tensor_dim0/1=256, tile_dim0/1=256, stride0=256
  {
    unsigned lds_addr = (unsigned)(unsigned long long)(void*)Cl;
    unsigned long long ga = (unsigned long long)(const void*)C;
    u32x4 g0;
    g0.x = 1u;                                              // count=1, flags=0
    g0.y = lds_addr;                                        // lds_addr
    g0.z = (unsigned)(ga & 0xFFFFFFFFull);                  // global_addr lo
    g0.w = ((unsigned)(ga >> 32) & 0x01FFFFFFu)             // global_addr[56:32]
           | 0x80000000u;                                   // type=2 ("image")
    u32x8 g1;
    g1.s0 = 0x00020000u;               // data_size=2 (4B); no multicast/pad/iter
    g1.s1 = ((unsigned)LAP_N) << 16;   // tensor_dim0 = 256 (lo16)
    g1.s2 = ((unsigned)LAP_N) << 16;   // tensor_dim1 = 256 (lo16)
    g1.s3 = ((unsigned)LAP_N) << 16;   // tile_dim0   = 256
    g1.s4 = (unsigned)LAP_N;           // tile_dim1   = 256, tile_dim2 = 0
    g1.s5 = (unsigned)LAP_N;           // tensor_dim0_stride = 256 (lo32)
    g1.s6 = 0u;                        // stride hi / dim1_stride lo
    g1.s7 = 0u;                        // dim1_stride hi
    asm volatile("tensor_load_to_lds %0, %1" :: "s"(g0), "s"(g1) : "memory");
  }

  // Init duals & matching while the TDM streams 256KB into LDS.
  for (int j = lane; j < LAP_N1; j += 32) { u[j] = 0.0f; p[j] = 0; }

  float v[LAP_K];
  #pragma unroll
  for (int k = 0; k < LAP_K; ++k) v[k] = 0.0f;

  asm volatile("s_wait_tensorcnt 0x0" ::: "memory");

  for (int i = 1; i <= LAP_N; ++i) {
    if (lane == 0) p[0] = i;
    float minv[LAP_K];
    int   pj_r[LAP_K];                 // cached p[j] for columns we marked used
    #pragma unroll
    for (int k = 0; k < LAP_K; ++k) { minv[k] = LAP_BIG; pj_r[k] = 0; }
    #pragma unroll
    for (int k = 0; k < LAP_K; ++k) way[k * 32 + lane] = 0;
    unsigned usedMask = 0u;
    int j0 = 0;
    __builtin_amdgcn_wave_barrier();

    // ---- shortest augmenting path (Dijkstra with duals) ----
    while (true) {
      int i0 = p[j0];                       // uniform broadcast load
      if (i0 == 0) break;                   // j0 is a free column -> augment
      // mark j0 used on its owner lane; cache its matched row (p[j0] == i0)
      #pragma unroll
      for (int k = 0; k < LAP_K; ++k) {
        if (lane == (j0 & 31) && k == (j0 >> 5)) {
          usedMask |= (1u << k);
          pj_r[k] = i0;
        }
      }
      float ui0 = u[i0];
      const float* Crow = Cl + (i0 - 1) * LAP_N;

      // reduced-cost sweep over unused real columns (j in [1,256])
      // k==0: j=lane   -> valid iff lane!=0 ; k==8: j=256+lane -> lane==0 only
      #pragma unroll
      for (int k = 0; k < LAP_K; ++k) {
        const bool validCol = (k == 0) ? (lane != 0)
                            : (k == LAP_K - 1) ? (lane == 0) : true;
        int j = k * 32 + lane;
        bool isused = (usedMask >> k) & 1u;
        if (validCol && !isused) {
          float cur = Crow[j - 1] - ui0 - v[k];
          if (cur < minv[k]) { minv[k] = cur; way[j] = j0; }
        }
      }

      // wave-wide masked argmin; ties resolved to lowest j (matches argmin)
      float bv = LAP_BIG;
      int   bj = 0x7FFFFFFF;
      #pragma unroll
      for (int k = 0; k < LAP_K; ++k) {
        const bool validCol = (k == 0) ? (lane != 0)
                            : (k == LAP_K - 1) ? (lane == 0) : true;
        int j = k * 32 + lane;
        bool isused = (usedMask >> k) & 1u;
        float cand = (validCol && !isused) ? minv[k] : LAP_BIG;
        if (cand < bv || (cand == bv && j < bj)) { bv = cand; bj = j; }
      }
      #pragma unroll
      for (int off = 16; off > 0; off >>= 1) {
        float ov = __shfl_xor(bv, off, 32);
        int   oj = __shfl_xor(bj, off, 32);
        if (ov < bv || (ov == bv && oj < bj)) { bv = ov; bj = oj; }
      }
      float delta = bv;
      int   j1    = bj;

      // dual update: used cols (incl dummy j=0) -> v -= d, u[p[j]] += d ;
      // unused valid cols -> minv -= d
      #pragma unroll
      for (int k = 0; k < LAP_K; ++k) {
        const bool validAny = (k == LAP_K - 1) ? (lane == 0) : true; // j<=256
        if (validAny) {
          if ((usedMask >> k) & 1u) {
            v[k] -= delta;
            u[pj_r[k]] += delta;           // distinct rows -> no LDS conflict
          } else {
            minv[k] -= delta;
          }
        }
      }
      j0 = j1;
    }

    // ---- augment along the alternating path back to dummy column 0 ----
    if (lane == 0) {
      int j = j0;
      while (j != 0) {
        int jw = way[j];
        p[j] = p[jw];
        j = jw;
      }
    }
    __builtin_amdgcn_wave_barrier();
  }

  // out[(p[c+1]-1) * N + c] = 1  (out already zeroed by lap_zero_out)
  for (int c = lane; c < LAP_N; c += 32) {
    int r = p[c + 1] - 1;
    out[r * LAP_N + c] = 1.0f;
  }
}

// ---------------------------------------------------------------------------
// Zero the 256x256 f32 output across the whole chip (bandwidth-trivial)
// ---------------------------------------------------------------------------
__global__ __launch_bounds__(256) void lap_zero_out(float* __restrict__ out) {
  int idx = blockIdx.x * blockDim.x + threadIdx.x;   // 64*256 = 16384 float4s
  float4 z = make_float4(0.f, 0.f, 0.f, 0.f);
  reinterpret_cast<float4*>(out)[idx] = z;
}

// ---------------------------------------------------------------------------
extern "C" void kernel_launch(void* const* d_in, const int* in_sizes, int n_in,
                              void* d_out, int out_size, void* d_ws, size_t ws_size,
                              hipStream_t stream) {
  (void)in_sizes; (void)n_in; (void)d_ws; (void)ws_size; (void)out_size;
  const float* C   = (const float*)d_in[0];   // unaries: [256,256] f32
  float*       out = (float*)d_out;           // [256,256] f32 0/1 labelling

  lap_zero_out<<<dim3(64), dim3(256), 0, stream>>>(out);
  lap_solve<<<dim3(1), dim3(32), SMEM_BYTES, stream>>>(C, out);
}